// Attn_47768626266275
// MI455X (gfx1250) — compile-verified
//
#include <hip/hip_runtime.h>
#include <hip/hip_bf16.h>

#define HID 1024
#define BATCH 32
#define SEQ 2048

typedef __attribute__((ext_vector_type(2))) float v2f;
typedef __attribute__((ext_vector_type(4))) float f4;
typedef __attribute__((ext_vector_type(8))) float v8f;

// ---------------------------------------------------------------------------
// Kernel 1: V[b,h] = sum_o hidden[b,o] * W[o,h]   (32x1024 = hidden @ W)
// f32 WMMA 16x16x4: M = batch tile, N = h tile, K = o.
// One wave per 16x16 output tile; 2 M-tiles x 64 N-tiles = 128 tiles.
// A-frag (16x4 f32, 2 VGPRs/lane): lanes 0-15 hold M=lane, K={0,1};
//                                  lanes 16-31 hold M=lane-16, K={2,3}.
// B-frag (4x16 f32, 2 VGPRs/lane): lanes 0-15 hold N=lane, K={0,1};
//                                  lanes 16-31 hold N=lane-16, K={2,3}.
// C/D (16x16 f32, 8 VGPRs): vgpr i holds M=i (lanes 0-15) / M=8+i (lanes 16-31), N=lane%16.
// ---------------------------------------------------------------------------
__global__ void wmma_v_kernel(const float* __restrict__ hidden,
                              const float* __restrict__ W,
                              float* __restrict__ V) {
    const int wave = threadIdx.x >> 5;
    const int lane = threadIdx.x & 31;
    const int tile = blockIdx.x * 8 + wave;      // 0..127
    const int m0 = (tile & 1) * 16;              // batch tile base (0 or 16)
    const int n0 = (tile >> 1) * 16;             // h tile base (0..1008)

    const int row   = lane & 15;                 // M (for A) / N (for B)
    const int khalf = (lane >> 4) << 1;          // 0 or 2

    const float* arow = hidden + (m0 + row) * HID + khalf;   // A[M=row, K=khalf..]
    const float* bcol = W + n0 + row;                        // B[K, N=row] = W[o, n0+row]

    v8f acc = {};
#pragma unroll 4
    for (int k0 = 0; k0 < HID; k0 += 4) {
        v2f a = *(const v2f*)(arow + k0);                    // K = k0+khalf, k0+khalf+1 (contiguous)
        v2f b;
        b.x = bcol[(k0 + khalf) * HID];                      // W[k0+khalf,   n0+row]
        b.y = bcol[(k0 + khalf + 1) * HID];                  // W[k0+khalf+1, n0+row]
        acc = __builtin_amdgcn_wmma_f32_16x16x4_f32(
            /*neg_a=*/false, a, /*neg_b=*/false, b,
            /*c_mod=*/(short)0, acc, /*reuse_a=*/false, /*reuse_b=*/false);
    }

    const int mbase = (lane >> 4) << 3;          // 0 or 8
#pragma unroll
    for (int i = 0; i < 8; ++i) {
        V[(m0 + mbase + i) * HID + n0 + row] = acc[i];
    }
}

// ---------------------------------------------------------------------------
// Kernel 1b: c[b] = hidden[b,:] . bias
// ---------------------------------------------------------------------------
__global__ void cvec_kernel(const float* __restrict__ hidden,
                            const float* __restrict__ bias,
                            float* __restrict__ c) {
    __shared__ float red[256];
    const int b = blockIdx.x;
    float acc = 0.f;
    for (int i = threadIdx.x; i < HID; i += 256)
        acc += hidden[b * HID + i] * bias[i];
    red[threadIdx.x] = acc;
    __syncthreads();
    for (int s = 128; s > 0; s >>= 1) {
        if (threadIdx.x < s) red[threadIdx.x] += red[threadIdx.x + s];
        __syncthreads();
    }
    if (threadIdx.x == 0) c[b] = red[0];
}

// ---------------------------------------------------------------------------
// Kernel 2: scores[b,s] = enc[b,s,:] . V[b,:] + c[b]
// Streams 256 MB of enc exactly once -> non-temporal loads.
// Block = 256 (8 waves); grid = (SEQ/8, BATCH). One wave per s.
// V[b,:] staged in LDS once per block.
// ---------------------------------------------------------------------------
__global__ void scores_kernel(const float* __restrict__ enc,
                              const float* __restrict__ V,
                              const float* __restrict__ c,
                              float* __restrict__ scores) {
    __shared__ float vs[HID];
    const int b = blockIdx.y;
    for (int i = threadIdx.x; i < HID; i += 256)
        vs[i] = V[b * HID + i];
    __syncthreads();

    const int wave = threadIdx.x >> 5;
    const int lane = threadIdx.x & 31;
    const int s = blockIdx.x * 8 + wave;

    const f4* erow = (const f4*)(enc + ((size_t)b * SEQ + s) * HID);
    float acc = 0.f;
#pragma unroll
    for (int i = 0; i < 8; ++i) {
        const int v = i * 32 + lane;                     // float4 index within row
        f4 e = __builtin_nontemporal_load(erow + v);     // streamed once: NT hint
        const int h = v * 4;
        acc += e.x * vs[h] + e.y * vs[h + 1] + e.z * vs[h + 2] + e.w * vs[h + 3];
    }
    // wave32 butterfly reduction
#pragma unroll
    for (int off = 16; off > 0; off >>= 1)
        acc += __shfl_xor(acc, off, 32);
    if (lane == 0)
        scores[b * SEQ + s] = acc + c[b];
}

// ---------------------------------------------------------------------------
// Kernel 3: out[b,0,s] = softmax over s of scores[b,:]
// One block per batch row; 2048 elements, 256 threads.
// ---------------------------------------------------------------------------
__global__ void softmax_kernel(const float* __restrict__ scores,
                               float* __restrict__ out) {
    __shared__ float red[256];
    const int b = blockIdx.x;
    const float* row = scores + b * SEQ;
    const int tid = threadIdx.x;

    float m = -__builtin_inff();
    for (int i = tid; i < SEQ; i += 256) m = fmaxf(m, row[i]);
    red[tid] = m;
    __syncthreads();
    for (int s = 128; s > 0; s >>= 1) {
        if (tid < s) red[tid] = fmaxf(red[tid], red[tid + s]);
        __syncthreads();
    }
    const float mx = red[0];
    __syncthreads();

    float sum = 0.f;
    for (int i = tid; i < SEQ; i += 256) sum += __expf(row[i] - mx);
    red[tid] = sum;
    __syncthreads();
    for (int s = 128; s > 0; s >>= 1) {
        if (tid < s) red[tid] += red[tid + s];
        __syncthreads();
    }
    const float inv = 1.0f / red[0];
    for (int i = tid; i < SEQ; i += 256)
        out[b * SEQ + i] = __expf(row[i] - mx) * inv;
}

// ---------------------------------------------------------------------------
extern "C" void kernel_launch(void* const* d_in, const int* in_sizes, int n_in,
                              void* d_out, int out_size, void* d_ws, size_t ws_size,
                              hipStream_t stream) {
    const float* hidden = (const float*)d_in[0];   // [32,1024]
    const float* enc    = (const float*)d_in[1];   // [32,2048,1024]
    const float* W      = (const float*)d_in[2];   // [1024,1024] (out,in)
    const float* bias   = (const float*)d_in[3];   // [1024]
    float* out = (float*)d_out;                    // [32,1,2048]

    float* ws     = (float*)d_ws;
    float* V      = ws;                            // 32*1024 floats
    float* cvec   = ws + BATCH * HID;              // 32 floats
    float* scores = ws + BATCH * HID + 32;         // 32*2048 floats

    // 1) V = hidden @ W  (WMMA f32): 128 tiles, 8 waves/block -> 16 blocks
    wmma_v_kernel<<<16, 256, 0, stream>>>(hidden, W, V);
    // 1b) c[b] = hidden[b] . bias
    cvec_kernel<<<BATCH, 256, 0, stream>>>(hidden, bias, cvec);
    // 2) scores = enc . V + c   (bandwidth-bound: 256 MB NT stream)
    scores_kernel<<<dim3(SEQ / 8, BATCH), 256, 0, stream>>>(enc, V, cvec, scores);
    // 3) softmax over seq axis
    softmax_kernel<<<BATCH, 256, 0, stream>>>(scores, out);
}